// MinDistanceGraspLoss_5282809774824
// MI455X (gfx1250) — compile-verified
//
#include <hip/hip_runtime.h>
#include <stdint.h>

#define TPB    128
#define NWAVE  (TPB / 32)
#define MAX_C  16
#define SCAN_T 256
#define RED_T  1024

__device__ __forceinline__ float smooth_l1(float x) {
  float a = fabsf(x);
  return (a < 1.0f) ? 0.5f * x * x : a - 0.5f;
}

// Issue one 16-byte async global->LDS copy (CDNA5 ASYNCcnt path).
__device__ __forceinline__ void async_b128(uint32_t lds_addr, const char* gaddr) {
  asm volatile("global_load_async_to_lds_b128 %0, %1, off"
               :: "v"(lds_addr), "v"(gaddr) : "memory");
}
__device__ __forceinline__ void async_b64(uint32_t lds_addr, const char* gaddr) {
  asm volatile("global_load_async_to_lds_b64 %0, %1, off"
               :: "v"(lds_addr), "v"(gaddr) : "memory");
}
__device__ __forceinline__ void wait_async0() {
  asm volatile("s_wait_asynccnt 0" ::: "memory");
}

// ---------------------------------------------------------------------------
// Kernel 1: per-block sums of target_counts (wave32 shuffle reduction)
// ---------------------------------------------------------------------------
__global__ __launch_bounds__(TPB) void k_block_sums(const int* __restrict__ counts,
                                                    int* __restrict__ bsum, int n) {
  __shared__ int sw[NWAVE];
  const int t = threadIdx.x;
  const int i = blockIdx.x * TPB + t;
  int c = (i < n) ? counts[i] : 0;
  for (int off = 16; off > 0; off >>= 1) c += __shfl_down(c, off, 32);
  if ((t & 31) == 0) sw[t >> 5] = c;
  __syncthreads();
  if (t == 0) {
    int s = 0;
    for (int w = 0; w < NWAVE; ++w) s += sw[w];
    bsum[blockIdx.x] = s;
  }
}

// ---------------------------------------------------------------------------
// Kernel 2: single-block exclusive scan of block sums -> per-block row base
// ---------------------------------------------------------------------------
__global__ __launch_bounds__(SCAN_T) void k_scan(const int* __restrict__ bsum,
                                                 int* __restrict__ bofs, int nb) {
  __shared__ int tot[SCAN_T];
  int t = threadIdx.x;
  int per = (nb + SCAN_T - 1) / SCAN_T;
  int base = t * per;
  int s = 0;
  for (int k = 0; k < per; ++k) {
    int idx = base + k;
    s += (idx < nb) ? bsum[idx] : 0;
  }
  tot[t] = s;
  __syncthreads();
  for (int off = 1; off < SCAN_T; off <<= 1) {
    int v = (t >= off) ? tot[t - off] : 0;
    __syncthreads();
    tot[t] += v;
    __syncthreads();
  }
  int running = (t == 0) ? 0 : tot[t - 1];
  for (int k = 0; k < per; ++k) {
    int idx = base + k;
    if (idx < nb) { bofs[idx] = running; running += bsum[idx]; }
  }
}

// ---------------------------------------------------------------------------
// Kernel 3: main — async-stage pred + target slabs to LDS, ragged argmin,
//           fused smooth-L1/L1 loss, deterministic block reduction
// ---------------------------------------------------------------------------
__global__ __launch_bounds__(TPB) void k_main(const float* __restrict__ pred,
                                              const float* __restrict__ tgt,
                                              const int* __restrict__ counts,
                                              const int* __restrict__ bofs,
                                              float* __restrict__ bloss, int n) {
  __shared__ float s_tgt[TPB * MAX_C * 6];   // 49152 B target slab
  __shared__ float s_pred[TPB * 6];          //  3072 B pred slab
  __shared__ int   s_wsum[NWAVE];
  __shared__ float s_red[NWAVE];

  const int t    = threadIdx.x;
  const int lane = t & 31;
  const int w    = t >> 5;
  const int bstart = blockIdx.x * TPB;
  const int i    = bstart + t;
  const int c    = (i < n) ? counts[i] : 0;

  // ---- wave32 shuffle inclusive scan of counts, combined across 4 waves ---
  int incl = c;
  for (int off = 1; off < 32; off <<= 1) {
    int u = __shfl_up(incl, off, 32);
    if (lane >= off) incl += u;
  }
  if (lane == 31) s_wsum[w] = incl;
  __syncthreads();
  int wbase = 0;
  for (int k = 0; k < w; ++k) wbase += s_wsum[k];
  int total = 0;
  for (int k = 0; k < NWAVE; ++k) total += s_wsum[k];
  const int lstart = wbase + incl - c;       // exclusive offset within slab

  // ---- async copy target slab [base, base+total) rows into LDS ------------
  const int   base   = bofs[blockIdx.x];
  const char* gt_ptr = (const char*)tgt + (size_t)base * 24u;
  const uint32_t l_tgt = (uint32_t)(uintptr_t)&s_tgt[0];
  const int tbytes = total * 24;
  const int tch    = tbytes >> 4;
  for (int k = t; k < tch; k += TPB)
    async_b128(l_tgt + (uint32_t)(k << 4), gt_ptr + ((size_t)k << 4));
  if ((tbytes & 15) && t == 0)               // exact 8-byte tail, no overread
    async_b64(l_tgt + (uint32_t)(tch << 4), gt_ptr + ((size_t)tch << 4));

  // ---- async copy this block's pred slab (contiguous, 16B-aligned) --------
  const int valid  = min(TPB, n - bstart);   // rows in this block
  const char* gp_ptr = (const char*)pred + (size_t)bstart * 24u;
  const uint32_t l_pred = (uint32_t)(uintptr_t)&s_pred[0];
  const int pbytes = valid * 24;
  const int pch    = pbytes >> 4;
  for (int k = t; k < pch; k += TPB)
    async_b128(l_pred + (uint32_t)(k << 4), gp_ptr + ((size_t)k << 4));
  if ((pbytes & 15) && t == 0)
    async_b64(l_pred + (uint32_t)(pch << 4), gp_ptr + ((size_t)pch << 4));

  wait_async0();
  __syncthreads();

  // ---- per-thread prediction from LDS (stride-6: conflict-free in wave32) -
  const float p0 = s_pred[t * 6 + 0];
  const float p1 = s_pred[t * 6 + 1];
  const float p2 = s_pred[t * 6 + 2];
  const float p3 = s_pred[t * 6 + 3];
  const float p4 = s_pred[t * 6 + 4];

  // ---- ragged argmin over this prediction's <=16 target rows --------------
  float bestD = 3.402823466e+38f;
  float g0 = 0.f, g1 = 0.f, g2 = 0.f, g3 = 0.f, g4 = 0.f;
  for (int j = 0; j < c; ++j) {
    const float* r = &s_tgt[(lstart + j) * 6];
    float d0 = p0 - r[0], d1 = p1 - r[1];
    float d2 = p2 - r[2], d3 = p3 - r[3];
    float d4 = p4 - r[4];
    float d = d0 * d0 + d1 * d1 + d2 * d2 + d3 * d3 + d4 * d4;
    if (d < bestD) {                         // strict '<' => first occurrence
      bestD = d;
      g0 = r[0]; g1 = r[1]; g2 = r[2]; g3 = r[3]; g4 = r[4];
    }
  }

  float loss = 0.0f;
  if (c > 0) {
    loss = smooth_l1(p0 - g0) + smooth_l1(p1 - g1)
         + fabsf(p2 - g2) + fabsf(p3 - g3)
         + smooth_l1(p4 - g4);
  }

  // ---- deterministic block reduction: wave32 shuffles, then LDS -----------
  for (int off = 16; off > 0; off >>= 1)
    loss += __shfl_down(loss, off, 32);
  if (lane == 0) s_red[w] = loss;
  __syncthreads();
  if (t == 0) {
    float s = 0.0f;
    for (int k = 0; k < NWAVE; ++k) s += s_red[k];
    bloss[blockIdx.x] = s;
  }
}

// ---------------------------------------------------------------------------
// Kernel 4: deterministic final reduction + mean
// ---------------------------------------------------------------------------
__global__ __launch_bounds__(RED_T) void k_reduce(const float* __restrict__ bloss,
                                                  int nb, float* __restrict__ out,
                                                  float inv_n) {
  __shared__ float sh[RED_T];
  int t = threadIdx.x;
  int per = (nb + RED_T - 1) / RED_T;
  float s = 0.0f;
  for (int k = 0; k < per; ++k) {
    int idx = t * per + k;
    if (idx < nb) s += bloss[idx];
  }
  sh[t] = s;
  __syncthreads();
  for (int st = RED_T / 2; st > 0; st >>= 1) {
    if (t < st) sh[t] += sh[t + st];
    __syncthreads();
  }
  if (t == 0) out[0] = sh[0] * inv_n;
}

// ---------------------------------------------------------------------------
extern "C" void kernel_launch(void* const* d_in, const int* in_sizes, int n_in,
                              void* d_out, int out_size, void* d_ws, size_t ws_size,
                              hipStream_t stream) {
  const float* pred   = (const float*)d_in[0];
  const float* tgt    = (const float*)d_in[1];
  const int*   counts = (const int*)d_in[2];

  const int n  = in_sizes[0] / 6;            // N predictions
  const int nb = (n + TPB - 1) / TPB;        // blocks

  int*   bsum  = (int*)d_ws;                 // nb ints
  int*   bofs  = bsum + nb;                  // nb ints
  float* bloss = (float*)(bofs + nb);        // nb floats

  k_block_sums<<<nb, TPB, 0, stream>>>(counts, bsum, n);
  k_scan<<<1, SCAN_T, 0, stream>>>(bsum, bofs, nb);
  k_main<<<nb, TPB, 0, stream>>>(pred, tgt, counts, bofs, bloss, n);
  k_reduce<<<1, RED_T, 0, stream>>>(bloss, nb, (float*)d_out, 1.0f / (float)n);
}